// PositionalEncoding_9320079032502
// MI455X (gfx1250) — compile-verified
//
#include <hip/hip_runtime.h>
#include <hip/hip_bf16.h>

typedef __attribute__((ext_vector_type(2))) float        v2f;
typedef __attribute__((ext_vector_type(8))) float        v8f;
typedef __attribute__((ext_vector_type(4))) unsigned int v4u;
typedef __attribute__((ext_vector_type(4))) int          v4i;
typedef __attribute__((ext_vector_type(8))) int          v8i;

#define N_ATOMS   1000000
#define N_CLIQUE  500000
#define N_EDGES   2000000
#define HID       128
#define PEDIM     32
#define WPB       4   // waves per block (128 threads)

__device__ __forceinline__ v8f wmma4(v2f a, v2f b, v8f c) {
    // D = A(16x4 f32) * B(4x16 f32) + C(16x16 f32), full fp32 precision
    return __builtin_amdgcn_wmma_f32_16x16x4_f32(
        /*neg_a=*/false, a, /*neg_b=*/false, b,
        /*c_mod=*/(short)0, c, /*reuse_a=*/false, /*reuse_b=*/false);
}

__device__ __forceinline__ float nan0(float v) { return (v != v) ? 0.0f : v; }
__device__ __forceinline__ int   iclamp(int v, int lo, int hi) {
    return v < lo ? lo : (v > hi ? hi : v);
}

// ---------------------------------------------------------------------------
// Kernel 0: swizzle a KxN weight matrix into WMMA B-fragment order so each
// lane's (B[k][n], B[k+1][n]) pair is one contiguous 8-byte load.
//   Wf[(kc*2 + p)*N + n] = { W[(4kc+2p)*N + n], W[(4kc+2p+1)*N + n] }
// ---------------------------------------------------------------------------
__global__ void weight_swizzle_kernel(const float* __restrict__ W,
                                      v2f* __restrict__ Wf, int K, int N) {
    int i = blockIdx.x * blockDim.x + threadIdx.x;
    int total = (K / 4) * 2 * N;
    if (i >= total) return;
    int n  = i % N;
    int t  = i / N;          // kc*2 + p
    int p  = t & 1;
    int kc = t >> 1;
    int k  = kc * 4 + 2 * p;
    v2f v;
    v.x = W[k * N + n];
    v.y = W[(k + 1) * N + n];
    Wf[i] = v;
}

// ---------------------------------------------------------------------------
// Kernel 1: zero d_out (scatter accumulator target) and edge counts in ws
// ---------------------------------------------------------------------------
__global__ void pe_zero_kernel(float* __restrict__ out, float* __restrict__ cnt,
                               int n_out, int n_cnt) {
    int i = blockIdx.x * blockDim.x + threadIdx.x;
    int stride = gridDim.x * blockDim.x;
    for (int k = i; k < n_out; k += stride) out[k] = 0.0f;
    for (int k = i; k < n_cnt; k += stride) cnt[k] = 0.0f;
}

// ---------------------------------------------------------------------------
// Kernel 2: fused pe = nan0(graph_lpe[row]) @ lpe_w + lpe_b per EDGE tile,
// scatter-add into out[col, 0:64] with fp32 global atomics; count atomics.
// One wave = 16 edges. WMMA f32 16x16x4, K=32 (8 steps), N=64 (4 tiles).
// ---------------------------------------------------------------------------
__global__ void edge_scatter_kernel(const float* __restrict__ graph_lpe,
                                    const int*   __restrict__ row,
                                    const int*   __restrict__ col,
                                    const v2f*   __restrict__ lpe_wf, // swizzled [32x64]
                                    const float* __restrict__ lpe_b,  // [64]
                                    float* __restrict__ out,          // [N_CLIQUE][128]
                                    float* __restrict__ cnt,          // [N_CLIQUE]
                                    int n_edges, int n_atoms) {
    const int wave = threadIdx.x >> 5;
    const int lane = threadIdx.x & 31;
    const int hl   = lane >> 4;     // lane half: 0 (0-15), 1 (16-31)
    const int hm   = lane & 15;     // m-index for A rows / n-index for B,C
    const int e0   = (blockIdx.x * WPB + wave) * 16;
    if (e0 >= n_edges) return;      // whole-wave exit (before any WMMA)

    // ---- A fragments: 16 gathered rows of graph_lpe, NaN-masked ----
    int ea = e0 + hm; if (ea > n_edges - 1) ea = n_edges - 1;
    int ra = iclamp(row[ea], 0, n_atoms - 1);
    const float* Arow = graph_lpe + ra * PEDIM;
    v2f afr[8];
#pragma unroll
    for (int kc = 0; kc < 8; ++kc) {
        int k = kc * 4 + 2 * hl;
        afr[kc].x = nan0(Arow[k + 0]);
        afr[kc].y = nan0(Arow[k + 1]);
    }

    // ---- destination clique ids for the 8 M-rows this lane owns in C ----
    int cm[8];
#pragma unroll
    for (int j = 0; j < 8; ++j) {
        int e = e0 + j + 8 * hl; if (e > n_edges - 1) e = n_edges - 1;
        cm[j] = col[e];
    }

    const v2f* Bp = lpe_wf + hl * 64 + hm;   // + nt*16 below, stride 2*64 per kc
#pragma unroll
    for (int nt = 0; nt < 4; ++nt) {
        v8f acc = {};
#pragma unroll
        for (int kc = 0; kc < 8; ++kc) {
            v2f b = Bp[kc * 128 + nt * 16];  // single global_load_b64
            acc = wmma4(afr[kc], b, acc);
        }
        float bias = lpe_b[nt * 16 + hm];
#pragma unroll
        for (int j = 0; j < 8; ++j) {
            int e = e0 + j + 8 * hl;
            if (e < n_edges)
                unsafeAtomicAdd(&out[cm[j] * HID + nt * 16 + hm], acc[j] + bias);
        }
    }
    // ---- edge counts (lanes 0..15 cover the 16 edges of this tile) ----
    if (hl == 0) {
        int e = e0 + hm;
        if (e < n_edges) unsafeAtomicAdd(&cnt[col[e]], 1.0f);
    }
}

// ---------------------------------------------------------------------------
// Kernel 3: per 16-clique tile:
//   A1 = deg_emb[tree_degree[r]]               (TDM gather-mode load to LDS)
//   h  = relu(A1 @ W1 + b1)                    (WMMA, K=128)
//   x  = (x_clique + h) @ W2 + b2              (WMMA, K=128)
//   t  = nan0(tree_lpe) @ Wt + bt              (WMMA, K=32, fused into upper cols)
//   out[:, :64]  = x[:, :64] + sums/max(cnt,1) (sums pre-accumulated in out)
//   out[:, 64:]  = x[:, 64:] + t
// ---------------------------------------------------------------------------
__global__ void clique_kernel(const float* __restrict__ x_clique,
                              const float* __restrict__ tree_lpe,
                              const int*   __restrict__ tree_degree,
                              const float* __restrict__ deg_emb,   // [100][128]
                              const v2f*   __restrict__ W1f,       // swizzled [128x128]
                              const float* __restrict__ b1,        // [128]
                              const v2f*   __restrict__ W2f,       // swizzled [128x128]
                              const float* __restrict__ b2,        // [128]
                              const v2f*   __restrict__ Wtf,       // swizzled [32x64]
                              const float* __restrict__ bt,        // [64]
                              float* __restrict__ out,             // [N_CLIQUE][128]
                              const float* __restrict__ cnt,
                              int n_clique) {
    __shared__ __align__(16) float lds[WPB][16][HID];   // 32 KB staging tile
    const int wave = threadIdx.x >> 5;
    const int lane = threadIdx.x & 31;
    const int hl   = lane >> 4;
    const int hm   = lane & 15;
    const int rowBase = (blockIdx.x * WPB + wave) * 16;

    // ---- TDM gather: stage A1 = deg_emb[tree_degree[r]] rows into LDS ----
    // D# gather mode, 16-bit indices, 16 rows of 128 fp32 each, packed
    // back-to-back at lds[wave][0][0] (matches the [16][128] tile layout).
    {
        int idxw[8];
#pragma unroll
        for (int p = 0; p < 8; ++p) {
            int r0 = rowBase + 2 * p;     if (r0 > n_clique - 1) r0 = n_clique - 1;
            int r1 = rowBase + 2 * p + 1; if (r1 > n_clique - 1) r1 = n_clique - 1;
            unsigned int d0 = (unsigned int)iclamp(tree_degree[r0], 0, 99);
            unsigned int d1 = (unsigned int)iclamp(tree_degree[r1], 0, 99);
            idxw[p] = __builtin_amdgcn_readfirstlane((int)(d0 | (d1 << 16)));
        }
        unsigned long long ga = (unsigned long long)(uintptr_t)deg_emb;
        unsigned int lds_addr = (unsigned int)(uintptr_t)&lds[wave][0][0];
        v4u g0;
        g0.x = 0x80000001u;                        // count=1, 16b idx, gather_mode=1
        g0.y = (unsigned int)__builtin_amdgcn_readfirstlane((int)lds_addr);
        g0.z = (unsigned int)ga;                   // global_addr[31:0]
        g0.w = ((unsigned int)(ga >> 32) & 0x01FFFFFFu) | 0x80000000u;  // addr[56:32], type=2
        v8i g1;
        g1[0] = (int)(2u << 16);        // data_size = 4 bytes
        g1[1] = (int)(128u << 16);      // tensor_dim0 = 128 (bits 63:48)
        g1[2] = (int)(100u << 16);      // tensor_dim1 = 100 rows (bits 95:80)
        g1[3] = (int)(128u << 16);      // tile_dim0 = 128 (bits 127:112)
        g1[4] = 16;                     // tile_dim1 = #valid gather indices
        g1[5] = 128;                    // tensor_dim0_stride = 128 elements
        g1[6] = 0;
        g1[7] = 0;
        v4i g2, g3;
        g2[0] = idxw[0]; g2[1] = idxw[1]; g2[2] = idxw[2]; g2[3] = idxw[3];
        g3[0] = idxw[4]; g3[1] = idxw[5]; g3[2] = idxw[6]; g3[3] = idxw[7];
        v8i g4 = {0, 0, 0, 0, 0, 0, 0, 0};   // trailing SGPR group unused (<=2D/gather)
        __builtin_amdgcn_tensor_load_to_lds(g0, g1, g2, g3, g4, 0);
    }

    // prefetch x_clique rows (consumed after GEMM1) while the TDM runs
    {
        int rp = rowBase + hm; if (rp > n_clique - 1) rp = n_clique - 1;
        const float* xp = x_clique + rp * HID + hl * 64;
        __builtin_prefetch(xp, 0, 3);
        __builtin_prefetch(xp + 32, 0, 3);
    }

    __builtin_amdgcn_s_wait_tensorcnt(0);
    __syncthreads();

    // ---- GEMM1: h = A1 @ W1 (K=128, 8 N-tiles held in regs) ----
    const v2f* B1p = W1f + hl * HID + hm;   // + nt*16, stride 2*HID per kc
    v8f acc[8];
#pragma unroll
    for (int nt = 0; nt < 8; ++nt) acc[nt] = (v8f){};
    for (int kc = 0; kc < 32; ++kc) {
        v2f a = *(const v2f*)&lds[wave][hm][kc * 4 + 2 * hl];
#pragma unroll
        for (int nt = 0; nt < 8; ++nt) {
            v2f b = B1p[kc * 256 + nt * 16];
            acc[nt] = wmma4(a, b, acc[nt]);
        }
    }
    __syncthreads();

    // ---- A2 = x_clique + relu(h + b1), restaged into LDS ----
#pragma unroll
    for (int nt = 0; nt < 8; ++nt) {
        float bb = b1[nt * 16 + hm];
#pragma unroll
        for (int j = 0; j < 8; ++j) {
            int m = j + 8 * hl;
            int r = rowBase + m; if (r > n_clique - 1) r = n_clique - 1;
            float h = acc[nt][j] + bb;
            h = h > 0.0f ? h : 0.0f;
            lds[wave][m][nt * 16 + hm] = x_clique[r * HID + nt * 16 + hm] + h;
        }
    }
    __syncthreads();

    // ---- GEMM2: x = A2 @ W2 (K=128) ----
    const v2f* B2p = W2f + hl * HID + hm;
#pragma unroll
    for (int nt = 0; nt < 8; ++nt) acc[nt] = (v8f){};
    for (int kc = 0; kc < 32; ++kc) {
        v2f a = *(const v2f*)&lds[wave][hm][kc * 4 + 2 * hl];
#pragma unroll
        for (int nt = 0; nt < 8; ++nt) {
            v2f b = B2p[kc * 256 + nt * 16];
            acc[nt] = wmma4(a, b, acc[nt]);
        }
    }

    // ---- fuse tpe = nan0(tree_lpe) @ Wt into upper 4 N-tiles (K=32) ----
    {
        int rA = rowBase + hm; if (rA > n_clique - 1) rA = n_clique - 1;
        const float* Trow = tree_lpe + rA * PEDIM;
        v2f tfr[8];
#pragma unroll
        for (int kc = 0; kc < 8; ++kc) {
            int k = kc * 4 + 2 * hl;
            tfr[kc].x = nan0(Trow[k + 0]);
            tfr[kc].y = nan0(Trow[k + 1]);
        }
        const v2f* Btp = Wtf + hl * 64 + hm;
#pragma unroll
        for (int nt = 4; nt < 8; ++nt) {
#pragma unroll
            for (int kc = 0; kc < 8; ++kc) {
                v2f b = Btp[kc * 128 + (nt - 4) * 16];
                acc[nt] = wmma4(tfr[kc], b, acc[nt]);
            }
        }
    }

    // ---- finalize + store (all WMMA done; predication safe now) ----
#pragma unroll
    for (int nt = 0; nt < 4; ++nt) {            // cols 0..63: + scatter-mean
        float bb = b2[nt * 16 + hm];
#pragma unroll
        for (int j = 0; j < 8; ++j) {
            int m = j + 8 * hl;
            int r = rowBase + m;
            if (r < n_clique) {
                int o = r * HID + nt * 16 + hm;
                float s = out[o];               // pre-accumulated edge sums
                float c = cnt[r]; c = c > 1.0f ? c : 1.0f;
                out[o] = acc[nt][j] + bb + s / c;
            }
        }
    }
#pragma unroll
    for (int nt = 4; nt < 8; ++nt) {            // cols 64..127: + tpe (+bt)
        float bb = b2[nt * 16 + hm] + bt[(nt - 4) * 16 + hm];
#pragma unroll
        for (int j = 0; j < 8; ++j) {
            int m = j + 8 * hl;
            int r = rowBase + m;
            if (r < n_clique)
                out[r * HID + nt * 16 + hm] = acc[nt][j] + bb;
        }
    }
}

// ---------------------------------------------------------------------------
extern "C" void kernel_launch(void* const* d_in, const int* in_sizes, int n_in,
                              void* d_out, int out_size, void* d_ws, size_t ws_size,
                              hipStream_t stream) {
    const float* x_clique    = (const float*)d_in[0];
    const float* tree_lpe    = (const float*)d_in[1];
    const float* graph_lpe   = (const float*)d_in[2];
    const int*   tree_degree = (const int*)  d_in[3];
    const int*   row         = (const int*)  d_in[4];
    const int*   col         = (const int*)  d_in[5];
    const float* deg_emb     = (const float*)d_in[6];
    const float* deg_lin_w   = (const float*)d_in[7];
    const float* deg_lin_b   = (const float*)d_in[8];
    const float* deg_merge_w = (const float*)d_in[9];
    const float* deg_merge_b = (const float*)d_in[10];
    const float* tree_lpe_w  = (const float*)d_in[11];
    const float* tree_lpe_b  = (const float*)d_in[12];
    const float* lpe_w       = (const float*)d_in[13];
    const float* lpe_b       = (const float*)d_in[14];

    float* out = (float*)d_out;

    // ---- workspace layout ----
    char* ws = (char*)d_ws;
    float* cnt  = (float*)ws;                           // 500000 floats (2 MB)
    size_t off  = ((size_t)N_CLIQUE * sizeof(float) + 255) & ~(size_t)255;
    v2f* W1f    = (v2f*)(ws + off);                     // 8192 v2f (64 KB)
    v2f* W2f    = (v2f*)(ws + off + 64 * 1024);         // 8192 v2f (64 KB)
    v2f* Wtf    = (v2f*)(ws + off + 128 * 1024);        // 1024 v2f (8 KB)
    v2f* Lpef   = (v2f*)(ws + off + 136 * 1024);        // 1024 v2f (8 KB)

    // 0) swizzle weights into WMMA B-fragment order (one b64 load per WMMA)
    weight_swizzle_kernel<<<(HID * HID / 2 + 255) / 256, 256, 0, stream>>>(
        deg_lin_w, W1f, HID, HID);
    weight_swizzle_kernel<<<(HID * HID / 2 + 255) / 256, 256, 0, stream>>>(
        deg_merge_w, W2f, HID, HID);
    weight_swizzle_kernel<<<(PEDIM * 64 / 2 + 255) / 256, 256, 0, stream>>>(
        tree_lpe_w, Wtf, PEDIM, 64);
    weight_swizzle_kernel<<<(PEDIM * 64 / 2 + 255) / 256, 256, 0, stream>>>(
        lpe_w, Lpef, PEDIM, 64);

    // 1) zero accumulators
    pe_zero_kernel<<<2048, 256, 0, stream>>>(out, cnt, N_CLIQUE * HID, N_CLIQUE);

    // 2) fused pe-compute + scatter-add (2M edges, 16 per wave)
    {
        int tiles  = (N_EDGES + 15) / 16;
        int blocks = (tiles + WPB - 1) / WPB;
        edge_scatter_kernel<<<blocks, 32 * WPB, 0, stream>>>(
            graph_lpe, row, col, Lpef, lpe_b, out, cnt, N_EDGES, N_ATOMS);
    }

    // 3) clique-side GEMMs + tpe + mean-merge (500K cliques, 16 per wave)
    {
        int tiles  = (N_CLIQUE + 15) / 16;
        int blocks = (tiles + WPB - 1) / WPB;
        clique_kernel<<<blocks, 32 * WPB, 0, stream>>>(
            x_clique, tree_lpe, tree_degree, deg_emb,
            W1f, deg_lin_b, W2f, deg_merge_b,
            Wtf, tree_lpe_b, out, cnt, N_CLIQUE);
    }
}